// EnhancedSeasonalModule_180388627307
// MI455X (gfx1250) — compile-verified
//
#include <hip/hip_runtime.h>

typedef __attribute__((ext_vector_type(2))) float v2f;
typedef __attribute__((ext_vector_type(8))) float v8f;

#define B_   16
#define S_   288
#define N_   170
#define C_   128
#define CL_  24
#define CHUNK 48
#define ROWS (CHUNK + 2)   // 50: halo row each side
#define ASTRIDE 132        // padded LDS row stride (floats)
#define LN_EPS 1e-5f

// Prep: ws[0..16383]  = W_c in WMMA-B fragment-packed layout:
//         Bpack[(k>>1)*256 + 2*d + (k&1)] = W_c[d*128 + k]
//       ws[16384..16511] = lin_b + b_c
__global__ void prep_kernel(const float* __restrict__ W,
                            const float* __restrict__ lin_b,
                            const float* __restrict__ b_c,
                            float* __restrict__ ws) {
  int i = blockIdx.x * blockDim.x + threadIdx.x;
  if (i < C_ * C_) {
    int d = i / C_, k = i % C_;
    ws[(k >> 1) * (2 * C_) + 2 * d + (k & 1)] = W[i];
  }
  if (i < C_) ws[C_ * C_ + i] = lin_b[i] + b_c[i];
}

__launch_bounds__(256)
__global__ void fused_kernel(const float* __restrict__ seasonal,
                             const int*   __restrict__ cycle_index,
                             const float* __restrict__ cycle_data,
                             const float* __restrict__ Wt_bias,
                             const float* __restrict__ conv_w,
                             const float* __restrict__ conv_b,
                             const float* __restrict__ ln_w,
                             const float* __restrict__ ln_b,
                             const float* __restrict__ gamma_p,
                             float* __restrict__ out) {
  __shared__ float As[64][ASTRIDE];     // combined = seasonal*cyc, K=128 per row
  __shared__ float T[ROWS][ASTRIDE];    // transformed (post-linear, pre-conv)

  const int chunk = blockIdx.x;
  const int n     = blockIdx.y;
  const int b     = blockIdx.z;
  const int s0    = chunk * CHUNK;
  const int tid   = threadIdx.x;
  const int lane  = tid & 31;
  const int w     = tid >> 5;          // wave id, 0..7
  const int half  = lane >> 4;         // 0: lanes 0-15, 1: lanes 16-31
  const int l16   = lane & 15;

  const int ci = ((cycle_index[b] % CL_) + CL_) % CL_;
  const float* __restrict__ Bpack = Wt_bias;
  const float* __restrict__ bias  = Wt_bias + C_ * C_;

  // ---- Stage A tile: As[r][c] = seasonal[b, s0-1+r, n, c] * cyc[(ci+s)%24, c]
  for (int i = tid; i < ROWS * (C_ / 4); i += 256) {
    int r  = i >> 5;
    int c4 = (i & 31) * 4;
    int s  = s0 - 1 + r;
    float4 v = make_float4(0.f, 0.f, 0.f, 0.f);
    if (s >= 0 && s < S_) {
      const float4 a = *(const float4*)(seasonal +
          ((size_t)((b * S_ + s) * N_ + n)) * C_ + c4);
      const float4 cy = *(const float4*)(cycle_data + ((ci + s) % CL_) * C_ + c4);
      v = make_float4(a.x * cy.x, a.y * cy.y, a.z * cy.z, a.w * cy.w);
    }
    As[r][c4 + 0] = v.x; As[r][c4 + 1] = v.y;
    As[r][c4 + 2] = v.z; As[r][c4 + 3] = v.w;
  }
  // zero the pad rows (read by WMMA fragments, results discarded)
  for (int i = tid; i < (64 - ROWS) * (C_ / 4); i += 256) {
    int r  = ROWS + (i >> 5);
    int c4 = (i & 31) * 4;
    As[r][c4 + 0] = 0.f; As[r][c4 + 1] = 0.f;
    As[r][c4 + 2] = 0.f; As[r][c4 + 3] = 0.f;
  }
  __syncthreads();

  // ---- GEMM: wave w owns rows [wm*16, +16), cols [wd*64, +64); 16x16x4 f32 WMMA
  const int wm = w >> 1;
  const int wd = w & 1;
  const int m  = wm * 16 + l16;        // A-matrix row for this lane
  v8f acc[4] = {};
#pragma unroll 4
  for (int kk = 0; kk < 32; ++kk) {
    const int kb = kk * 4 + 2 * half;  // lanes 0-15: K=k0,k0+1; 16-31: k0+2,k0+3
    const int kp = kk * 2 + half;      // packed-B row pair index (kb>>1)
    v2f a;
    a.x = As[m][kb];
    a.y = As[m][kb + 1];
    const float* __restrict__ brow = Bpack + kp * (2 * C_);
#pragma unroll
    for (int dt = 0; dt < 4; ++dt) {
      const int d = wd * 64 + dt * 16 + l16;
      v2f bf;
      const float2 bp = *(const float2*)(brow + 2 * d);  // one b64 load
      bf.x = bp.x;
      bf.y = bp.y;
      acc[dt] = __builtin_amdgcn_wmma_f32_16x16x4_f32(
          false, a, false, bf, (short)0, acc[dt], false, false);
    }
  }

  // ---- Bias add + store tile to LDS (zero rows outside [0,S): conv SAME pad)
#pragma unroll
  for (int dt = 0; dt < 4; ++dt) {
    const int d  = wd * 64 + dt * 16 + l16;
    const float bv = bias[d];
#pragma unroll
    for (int j = 0; j < 8; ++j) {
      int r = wm * 16 + j + 8 * half;
      if (r < ROWS) {
        int s = s0 - 1 + r;
        T[r][d] = (s >= 0 && s < S_) ? (acc[dt][j] + bv) : 0.f;
      }
    }
  }
  __syncthreads();

  // ---- Depthwise conv3 + exact GELU + LayerNorm(C) + residual
  const float gma = gamma_p[0];
  for (int r = 1 + w; r <= CHUNK; r += 8) {   // each wave: 6 output rows
    const int s = s0 - 1 + r;
    float g[4];
    float lsum = 0.f;
#pragma unroll
    for (int j = 0; j < 4; ++j) {
      int c = lane + j * 32;
      float y = conv_w[c * 3 + 0] * T[r - 1][c]
              + conv_w[c * 3 + 1] * T[r][c]
              + conv_w[c * 3 + 2] * T[r + 1][c]
              + conv_b[c];
      float e = 0.5f * y * (1.f + erff(y * 0.70710678118654752f));
      g[j] = e;
      lsum += e;
    }
#pragma unroll
    for (int o = 16; o > 0; o >>= 1) lsum += __shfl_xor(lsum, o, 32);
    const float mu = lsum * (1.f / 128.f);
    float lvar = 0.f;
#pragma unroll
    for (int j = 0; j < 4; ++j) { float dd = g[j] - mu; lvar += dd * dd; }
#pragma unroll
    for (int o = 16; o > 0; o >>= 1) lvar += __shfl_xor(lvar, o, 32);
    const float rinv = rsqrtf(lvar * (1.f / 128.f) + LN_EPS);

    const size_t base = ((size_t)((b * S_ + s) * N_ + n)) * C_;
#pragma unroll
    for (int j = 0; j < 4; ++j) {
      int c = lane + j * 32;
      float ln = (g[j] - mu) * rinv * ln_w[c] + ln_b[c];
      out[base + c] = seasonal[base + c] + gma * ln;
    }
  }
}

extern "C" void kernel_launch(void* const* d_in, const int* in_sizes, int n_in,
                              void* d_out, int out_size, void* d_ws, size_t ws_size,
                              hipStream_t stream) {
  const float* seasonal = (const float*)d_in[0];
  const int*   cyc_idx  = (const int*)  d_in[1];
  const float* cyc_data = (const float*)d_in[2];
  const float* W_c      = (const float*)d_in[3];
  const float* lin_b    = (const float*)d_in[4];
  const float* b_c      = (const float*)d_in[5];
  const float* conv_w   = (const float*)d_in[6];
  const float* conv_b   = (const float*)d_in[7];
  const float* ln_w     = (const float*)d_in[8];
  const float* ln_b     = (const float*)d_in[9];
  const float* gamma    = (const float*)d_in[10];
  float* out = (float*)d_out;
  float* ws  = (float*)d_ws;

  prep_kernel<<<64, 256, 0, stream>>>(W_c, lin_b, b_c, ws);

  dim3 grid(S_ / CHUNK, N_, B_);   // 6 x 170 x 16 = 16320 blocks
  fused_kernel<<<grid, 256, 0, stream>>>(seasonal, cyc_idx, cyc_data, ws,
                                         conv_w, conv_b, ln_w, ln_b, gamma, out);
}